// NGCF_65712999628849
// MI455X (gfx1250) — compile-verified
//
#include <hip/hip_runtime.h>

typedef __attribute__((ext_vector_type(2))) float v2f;
typedef __attribute__((ext_vector_type(8))) float v8f;

#define NEG_SLOPE 0.2f
#define L2_EPS 1e-12f

__device__ __forceinline__ void atomAddF(float* p, float v) {
    __hip_atomic_fetch_add(p, v, __ATOMIC_RELAXED, __HIP_MEMORY_SCOPE_AGENT);
}

// Stage W (64x64 row-major) into LDS pre-swizzled into WMMA B-fragment order:
// for k-step (k0=4*step), k-half hi, col n = nlo + 16*slab, the two K-consecutive
// weights W[k0+2*hi+0][n], W[k0+2*hi+1][n] are adjacent -> one ds_load_b64 per frag.
__device__ __forceinline__ void stage_W_swizzled(const float* __restrict__ W,
                                                 float* __restrict__ sWt,
                                                 int tid, int nthreads) {
    for (int idx = tid; idx < 64 * 64; idx += nthreads) {
        const int k = idx >> 6;       // coalesced global read: W[k][n]
        const int n = idx & 63;
        const int step = k >> 2;
        const int hi   = (k >> 1) & 1;
        const int half = k & 1;
        const int slab = n >> 4;
        const int nlo  = n & 15;
        sWt[(step * 2 + hi) * 128 + nlo * 8 + slab * 2 + half] = W[idx];
    }
}

// -------------------- degree count --------------------
__global__ void k_degrees(const int* __restrict__ src, const int* __restrict__ dst,
                          float* __restrict__ du, float* __restrict__ di, int E) {
    int e = blockIdx.x * blockDim.x + threadIdx.x;
    if (e < E) {
        atomAddF(&du[src[e]], 1.0f);
        atomAddF(&di[dst[e]], 1.0f);
    }
}

// -------------------- per-edge symmetric norm --------------------
__global__ void k_edge_norm(const int* __restrict__ src, const int* __restrict__ dst,
                            const float* __restrict__ du, const float* __restrict__ di,
                            float* __restrict__ nrm, int E) {
    int e = blockIdx.x * blockDim.x + threadIdx.x;
    if (e < E) {
        nrm[e] = rsqrtf(du[src[e]] * di[dst[e]]);
    }
}

// -------------------- node transform: Y = X @ W + b  (K=64, d=64) --------------------
// One wave per 16-row tile; 4 waves per block. fp32 WMMA 16x16x4.
__global__ __launch_bounds__(128) void k_node_xform(const float* __restrict__ X,
                                                    const float* __restrict__ W,
                                                    const float* __restrict__ b,
                                                    float* __restrict__ Y, int N) {
    __shared__ float sWt[64 * 64];
    stage_W_swizzled(W, sWt, threadIdx.x, blockDim.x);
    __syncthreads();

    const int wave = threadIdx.x >> 5;
    const int lane = threadIdx.x & 31;
    const int tile = blockIdx.x * 4 + wave;
    const int row0 = tile * 16;
    if (row0 >= N) return;  // wave-uniform

    const int m = lane & 15;
    const int hi = lane >> 4;
    const int khalf = hi * 2;
    int row = row0 + m;
    if (row >= N) row = N - 1;  // clamp (loads only)
    const float* xrow = X + (size_t)row * 64;
    const float* wbase = &sWt[hi * 128 + m * 8];

    v8f acc0 = {}, acc1 = {}, acc2 = {}, acc3 = {};
    for (int k0 = 0; k0 < 64; k0 += 4) {
        v2f a;
        a.x = xrow[k0 + khalf];
        a.y = xrow[k0 + khalf + 1];
        const float* wp = wbase + (k0 >> 2) * 256;
        v2f bf0 = *(const v2f*)(wp + 0);
        v2f bf1 = *(const v2f*)(wp + 2);
        v2f bf2 = *(const v2f*)(wp + 4);
        v2f bf3 = *(const v2f*)(wp + 6);
        acc0 = __builtin_amdgcn_wmma_f32_16x16x4_f32(false, a, false, bf0, (short)0, acc0, false, false);
        acc1 = __builtin_amdgcn_wmma_f32_16x16x4_f32(false, a, false, bf1, (short)0, acc1, false, false);
        acc2 = __builtin_amdgcn_wmma_f32_16x16x4_f32(false, a, false, bf2, (short)0, acc2, false, false);
        acc3 = __builtin_amdgcn_wmma_f32_16x16x4_f32(false, a, false, bf3, (short)0, acc3, false, false);
    }

    const int n0 = lane & 15;
    const float bb0 = b[n0], bb1 = b[n0 + 16], bb2 = b[n0 + 32], bb3 = b[n0 + 48];
#pragma unroll
    for (int r = 0; r < 8; ++r) {
        int mrow = row0 + r + 8 * hi;
        if (mrow >= N) continue;
        float* yrow = Y + (size_t)mrow * 64;
        yrow[n0]      = acc0[r] + bb0;
        yrow[n0 + 16] = acc1[r] + bb1;
        yrow[n0 + 32] = acc2[r] + bb2;
        yrow[n0 + 48] = acc3[r] + bb3;
    }
}

// -------------------- edge message-passing --------------------
// Per wave: 16 edges. cross = u[src]*i[dst]; mW2 = cross@W2 + b2 (WMMA);
// atomically scatter norm*u1[src]+mW2 -> accI[dst], norm*i1[dst]+mW2 -> accU[src].
__global__ __launch_bounds__(128) void k_edge_mp(
    const float* __restrict__ Ucur, const float* __restrict__ Icur,
    const float* __restrict__ U1, const float* __restrict__ I1,
    const float* __restrict__ W2, const float* __restrict__ b2,
    const int* __restrict__ src, const int* __restrict__ dst,
    const float* __restrict__ nrm,
    float* __restrict__ accU, float* __restrict__ accI, int E) {
    __shared__ float sWt[64 * 64];
    __shared__ float sCross[4][16 * 64];
    stage_W_swizzled(W2, sWt, threadIdx.x, blockDim.x);
    __syncthreads();

    const int wave = threadIdx.x >> 5;
    const int lane = threadIdx.x & 31;
    const int e0 = (blockIdx.x * 4 + wave) * 16;
    if (e0 >= E) return;  // wave-uniform

    const int m = lane & 15;
    const int hi = lane >> 4;
    int e_m = e0 + m;
    if (e_m >= E) e_m = E - 1;
    const int s_m = src[e_m];
    const int d_m = dst[e_m];
    const float nrm_m = nrm[e_m];

    // Build cross tile in LDS: lane handles edge (lane&15), columns hi*32..hi*32+31
    {
        const float* urow = Ucur + (size_t)s_m * 64 + hi * 32;
        const float* irow = Icur + (size_t)d_m * 64 + hi * 32;
        float* crow = &sCross[wave][m * 64 + hi * 32];
#pragma unroll
        for (int j = 0; j < 32; j += 4) {
            float4 uv = *(const float4*)(urow + j);
            float4 iv = *(const float4*)(irow + j);
            crow[j + 0] = uv.x * iv.x;
            crow[j + 1] = uv.y * iv.y;
            crow[j + 2] = uv.z * iv.z;
            crow[j + 3] = uv.w * iv.w;
        }
    }
    // LDS ops from the same wave execute in order; sCross slice is private per wave.

    const int khalf = hi * 2;
    const float* crow = &sCross[wave][m * 64];
    const float* wbase = &sWt[hi * 128 + m * 8];
    v8f acc0 = {}, acc1 = {}, acc2 = {}, acc3 = {};
    for (int k0 = 0; k0 < 64; k0 += 4) {
        v2f a = *(const v2f*)(crow + k0 + khalf);
        const float* wp = wbase + (k0 >> 2) * 256;
        v2f bf0 = *(const v2f*)(wp + 0);
        v2f bf1 = *(const v2f*)(wp + 2);
        v2f bf2 = *(const v2f*)(wp + 4);
        v2f bf3 = *(const v2f*)(wp + 6);
        acc0 = __builtin_amdgcn_wmma_f32_16x16x4_f32(false, a, false, bf0, (short)0, acc0, false, false);
        acc1 = __builtin_amdgcn_wmma_f32_16x16x4_f32(false, a, false, bf1, (short)0, acc1, false, false);
        acc2 = __builtin_amdgcn_wmma_f32_16x16x4_f32(false, a, false, bf2, (short)0, acc2, false, false);
        acc3 = __builtin_amdgcn_wmma_f32_16x16x4_f32(false, a, false, bf3, (short)0, acc3, false, false);
    }

    const int n0 = lane & 15;
    const float bb0 = b2[n0], bb1 = b2[n0 + 16], bb2 = b2[n0 + 32], bb3 = b2[n0 + 48];
#pragma unroll
    for (int r = 0; r < 8; ++r) {
        const int mm = r + 8 * hi;  // edge row within tile
        const int s = __shfl(s_m, mm);
        const int d = __shfl(d_m, mm);
        const float nr = __shfl(nrm_m, mm);
        const int e = e0 + mm;
        if (e >= E) continue;
        const float* u1r = U1 + (size_t)s * 64;
        const float* i1r = I1 + (size_t)d * 64;
        float* aIr = accI + (size_t)d * 64;
        float* aUr = accU + (size_t)s * 64;
        float v0 = acc0[r] + bb0;
        float v1 = acc1[r] + bb1;
        float v2 = acc2[r] + bb2;
        float v3 = acc3[r] + bb3;
        atomAddF(&aIr[n0],      nr * u1r[n0]      + v0);
        atomAddF(&aIr[n0 + 16], nr * u1r[n0 + 16] + v1);
        atomAddF(&aIr[n0 + 32], nr * u1r[n0 + 32] + v2);
        atomAddF(&aIr[n0 + 48], nr * u1r[n0 + 48] + v3);
        atomAddF(&aUr[n0],      nr * i1r[n0]      + v0);
        atomAddF(&aUr[n0 + 16], nr * i1r[n0 + 16] + v1);
        atomAddF(&aUr[n0 + 32], nr * i1r[n0 + 32] + v2);
        atomAddF(&aUr[n0 + 48], nr * i1r[n0 + 48] + v3);
    }
}

// -------------------- epilogue: leaky_relu + L2 row-normalize --------------------
// One wave handles 2 rows; each half-wave (16 lanes) owns a row, 4 cols/lane.
__global__ void k_epilogue(const float* __restrict__ acc, float* __restrict__ out, int N) {
    const int gwave = (blockIdx.x * blockDim.x + threadIdx.x) >> 5;
    const int lane = threadIdx.x & 31;
    const int row = gwave * 2 + (lane >> 4);
    if (row >= N) return;
    const int c0 = (lane & 15) * 4;
    float4 v = *(const float4*)(acc + (size_t)row * 64 + c0);
    v.x = v.x >= 0.0f ? v.x : NEG_SLOPE * v.x;
    v.y = v.y >= 0.0f ? v.y : NEG_SLOPE * v.y;
    v.z = v.z >= 0.0f ? v.z : NEG_SLOPE * v.z;
    v.w = v.w >= 0.0f ? v.w : NEG_SLOPE * v.w;
    float ss = v.x * v.x + v.y * v.y + v.z * v.z + v.w * v.w;
    ss += __shfl_xor(ss, 1);
    ss += __shfl_xor(ss, 2);
    ss += __shfl_xor(ss, 4);
    ss += __shfl_xor(ss, 8);
    const float inv = 1.0f / fmaxf(sqrtf(ss), L2_EPS);
    v.x *= inv; v.y *= inv; v.z *= inv; v.w *= inv;
    *(float4*)(out + (size_t)row * 64 + c0) = v;
}

// -------------------- final gather: concat [emb, L1, L2] rows --------------------
__global__ void k_gather(const float* __restrict__ u0, const float* __restrict__ uL1,
                         const float* __restrict__ uL2, const float* __restrict__ i0,
                         const float* __restrict__ iL1, const float* __restrict__ iL2,
                         const int* __restrict__ users, const int* __restrict__ pos,
                         const int* __restrict__ neg, float* __restrict__ out, int B) {
    const int t = blockIdx.x * blockDim.x + threadIdx.x;
    const int total = 3 * B * 192;
    if (t >= total) return;
    const int c = t % 192;
    const int rem = t / 192;
    const int bidx = rem % B;
    const int which = rem / B;
    const float *e0, *e1, *e2;
    int node;
    if (which == 0) { node = users[bidx]; e0 = u0; e1 = uL1; e2 = uL2; }
    else if (which == 1) { node = pos[bidx]; e0 = i0; e1 = iL1; e2 = iL2; }
    else { node = neg[bidx]; e0 = i0; e1 = iL1; e2 = iL2; }
    float v;
    if (c < 64)       v = e0[(size_t)node * 64 + c];
    else if (c < 128) v = e1[(size_t)node * 64 + (c - 64)];
    else              v = e2[(size_t)node * 64 + (c - 128)];
    out[t] = v;
}

extern "C" void kernel_launch(void* const* d_in, const int* in_sizes, int n_in,
                              void* d_out, int out_size, void* d_ws, size_t ws_size,
                              hipStream_t stream) {
    const float* u0   = (const float*)d_in[0];
    const float* i0   = (const float*)d_in[1];
    const float* W1_0 = (const float*)d_in[2];
    const float* b1_0 = (const float*)d_in[3];
    const float* W2_0 = (const float*)d_in[4];
    const float* b2_0 = (const float*)d_in[5];
    const float* W1_1 = (const float*)d_in[6];
    const float* b1_1 = (const float*)d_in[7];
    const float* W2_1 = (const float*)d_in[8];
    const float* b2_1 = (const float*)d_in[9];
    const int* edge  = (const int*)d_in[10];
    const int* users = (const int*)d_in[11];
    const int* pos   = (const int*)d_in[12];
    const int* neg   = (const int*)d_in[13];

    const int NU = in_sizes[0] / 64;
    const int NI = in_sizes[1] / 64;
    const int E  = in_sizes[10] / 2;
    const int B  = in_sizes[11];
    const int* src = edge;
    const int* dst = edge + E;

    float* ws = (float*)d_ws;
    size_t off = 0;
    float* du   = ws + off; off += (size_t)NU;
    float* di   = ws + off; off += (size_t)NI;
    float* nrm  = ws + off; off += (size_t)E;
    float* u1b  = ws + off; off += (size_t)NU * 64;
    float* i1b  = ws + off; off += (size_t)NI * 64;
    float* accU = ws + off; off += (size_t)NU * 64;
    float* accI = ws + off; off += (size_t)NI * 64;
    float* uL1  = ws + off; off += (size_t)NU * 64;
    float* iL1  = ws + off; off += (size_t)NI * 64;
    float* uL2  = ws + off; off += (size_t)NU * 64;
    float* iL2  = ws + off; off += (size_t)NI * 64;

    // degrees + edge norms
    hipMemsetAsync(du, 0, (size_t)NU * sizeof(float), stream);
    hipMemsetAsync(di, 0, (size_t)NI * sizeof(float), stream);
    k_degrees<<<(E + 255) / 256, 256, 0, stream>>>(src, dst, du, di, E);
    k_edge_norm<<<(E + 255) / 256, 256, 0, stream>>>(src, dst, du, di, nrm, E);

    const int tU = (NU + 15) / 16, tI = (NI + 15) / 16, tE = (E + 15) / 16;

    // ---- layer 1 ----
    k_node_xform<<<(tU + 3) / 4, 128, 0, stream>>>(u0, W1_0, b1_0, u1b, NU);
    k_node_xform<<<(tI + 3) / 4, 128, 0, stream>>>(i0, W1_0, b1_0, i1b, NI);
    hipMemsetAsync(accU, 0, (size_t)NU * 64 * sizeof(float), stream);
    hipMemsetAsync(accI, 0, (size_t)NI * 64 * sizeof(float), stream);
    k_edge_mp<<<(tE + 3) / 4, 128, 0, stream>>>(u0, i0, u1b, i1b, W2_0, b2_0,
                                                src, dst, nrm, accU, accI, E);
    k_epilogue<<<(NU + 15) / 16, 256, 0, stream>>>(accU, uL1, NU);
    k_epilogue<<<(NI + 15) / 16, 256, 0, stream>>>(accI, iL1, NI);

    // ---- layer 2 ----
    k_node_xform<<<(tU + 3) / 4, 128, 0, stream>>>(uL1, W1_1, b1_1, u1b, NU);
    k_node_xform<<<(tI + 3) / 4, 128, 0, stream>>>(iL1, W1_1, b1_1, i1b, NI);
    hipMemsetAsync(accU, 0, (size_t)NU * 64 * sizeof(float), stream);
    hipMemsetAsync(accI, 0, (size_t)NI * 64 * sizeof(float), stream);
    k_edge_mp<<<(tE + 3) / 4, 128, 0, stream>>>(uL1, iL1, u1b, i1b, W2_1, b2_1,
                                                src, dst, nrm, accU, accI, E);
    k_epilogue<<<(NU + 15) / 16, 256, 0, stream>>>(accU, uL2, NU);
    k_epilogue<<<(NI + 15) / 16, 256, 0, stream>>>(accI, iL2, NI);

    // ---- final gather ----
    const int total = 3 * B * 192;
    k_gather<<<(total + 255) / 256, 256, 0, stream>>>(u0, uL1, uL2, i0, iL1, iL2,
                                                      users, pos, neg, (float*)d_out, B);
}